// SupportLowRankRNN_withMask_58145267253495
// MI455X (gfx1250) — compile-verified
//
#include <hip/hip_runtime.h>

#define ALPHA_C 0.2f
#define NOISE_C 0.05f
#define I_DIM 3
#define H_DIM 2048
#define O_DIM 2
#define R_DIM 2
#define S_DIM 4
#define G_DIM 7
#define B_DIM 32
#define T_DIM 512

typedef float v2f __attribute__((ext_vector_type(2)));
typedef float v4f __attribute__((ext_vector_type(4)));
typedef float v8f __attribute__((ext_vector_type(8)));

// d_ws layout, in units of H_DIM floats:
//  0..2 : wi_full rows i=0..2            (I,H)
//  3..4 : n_rec columns r=0,1            (H,R)
//  5..6 : m_rec columns r=0,1            (H,R)
//  7..8 : wo_full columns o=0,1          (H,O)
//  9    : bias_full                      (H,)
// 10    : h0_full                        (H,)
// 11    : zero column (B-fragment filler for WMMA lanes n>=2)

__global__ __launch_bounds__(256) void build_proxies(
    const float* __restrict__ wi, const float* __restrict__ unitwi,
    const float* __restrict__ m,  const float* __restrict__ n,
    const float* __restrict__ unitm, const float* __restrict__ unitn,
    const float* __restrict__ wo, const float* __restrict__ h0,
    const float* __restrict__ unith0, const float* __restrict__ bias,
    const float* __restrict__ gb, const float* __restrict__ uv,
    const float* __restrict__ sup, float* __restrict__ ws)
{
  const int h = blockIdx.x * blockDim.x + threadIdx.x;
  if (h >= H_DIM) return;
  float g[G_DIM];
#pragma unroll
  for (int gg = 0; gg < G_DIM; ++gg) g[gg] = gb[gg * H_DIM + h];
  const float u = uv[h];
  float sp[S_DIM];
#pragma unroll
  for (int s = 0; s < S_DIM; ++s) sp[s] = sup[s * H_DIM + h];

#pragma unroll
  for (int i = 0; i < I_DIM; ++i) {          // wi_full
    float acc = 0.f;
#pragma unroll
    for (int s = 0; s < S_DIM; ++s) {
      float t = 0.f;
#pragma unroll
      for (int gg = 0; gg < G_DIM; ++gg) t += wi[(i * S_DIM + s) * G_DIM + gg] * g[gg];
      t += unitwi[i * S_DIM + s] * u;
      acc += sp[s] * t;
    }
    ws[i * H_DIM + h] = acc;
  }
#pragma unroll
  for (int r = 0; r < R_DIM; ++r) {          // n_rec
    float acc = 0.f;
#pragma unroll
    for (int s = 0; s < S_DIM; ++s) {
      float t = 0.f;
#pragma unroll
      for (int gg = 0; gg < G_DIM; ++gg) t += n[(r * S_DIM + s) * G_DIM + gg] * g[gg];
      t += unitn[r * S_DIM + s] * u;
      acc += sp[s] * t;
    }
    ws[(3 + r) * H_DIM + h] = acc;
  }
#pragma unroll
  for (int r = 0; r < R_DIM; ++r) {          // m_rec
    float acc = 0.f;
#pragma unroll
    for (int s = 0; s < S_DIM; ++s) {
      float t = 0.f;
#pragma unroll
      for (int gg = 0; gg < G_DIM; ++gg) t += m[(r * S_DIM + s) * G_DIM + gg] * g[gg];
      t += unitm[r * S_DIM + s] * u;
      acc += sp[s] * t;
    }
    ws[(5 + r) * H_DIM + h] = acc;
  }
#pragma unroll
  for (int o = 0; o < O_DIM; ++o) {          // wo_full (no unit term)
    float acc = 0.f;
#pragma unroll
    for (int s = 0; s < S_DIM; ++s) {
      float t = 0.f;
#pragma unroll
      for (int gg = 0; gg < G_DIM; ++gg) t += wo[(o * S_DIM + s) * G_DIM + gg] * g[gg];
      acc += sp[s] * t;
    }
    ws[(7 + o) * H_DIM + h] = acc;
  }
  {                                          // bias_full
    float acc = 0.f;
#pragma unroll
    for (int s = 0; s < S_DIM; ++s) acc += sp[s] * bias[s] * u;
    ws[9 * H_DIM + h] = acc;
  }
  {                                          // h0_full
    float acc = 0.f;
#pragma unroll
    for (int s = 0; s < S_DIM; ++s) {
      float t = 0.f;
#pragma unroll
      for (int gg = 0; gg < G_DIM; ++gg) t += h0[s * G_DIM + gg] * g[gg];
      t += unith0[s] * u;
      acc += sp[s] * t;
    }
    ws[10 * H_DIM + h] = acc;
  }
  ws[11 * H_DIM + h] = 0.f;                  // zero filler column
}

// Branch-free tanh: tanh(x) = 1 - 2/(exp2(2*log2(e)*x) + 1).
// Raw v_exp_f32/v_rcp_f32; saturates to +/-1 via inf/0 with no control flow.
// Absolute error ~1e-7, which is what matters (tanh enters linearly).
__device__ __forceinline__ float tanh_fast(float x) {
  const float e = __builtin_amdgcn_exp2f(x * 2.885390081777927f);
  const float r = __builtin_amdgcn_rcpf(e + 1.0f);
  return __builtin_fmaf(-2.0f, r, 1.0f);
}

__device__ __forceinline__ v4f ld4(const float* p) { return *(const v4f*)p; }
__device__ __forceinline__ v4f ld4nt(const float* p) {
  return __builtin_nontemporal_load((const v4f*)p);
}
__device__ __forceinline__ v4f tanh4(v4f a) {
  v4f r;
  r.x = tanh_fast(a.x); r.y = tanh_fast(a.y);
  r.z = tanh_fast(a.z); r.w = tanh_fast(a.w);
  return r;
}
__device__ __forceinline__ float dot4(v4f a, v4f b) {
  return a.x * b.x + a.y * b.y + a.z * b.z + a.w * b.w;
}
__device__ __forceinline__ v4f upd(v4f h, v4f z, v4f bs, v4f m0, v4f m1,
                                   v4f w0, v4f w1, v4f w2,
                                   float s0, float s1, float x0, float x1, float x2)
{
  v4f rec = s0 * m0 + s1 * m1 + x0 * w0 + x1 * w1 + x2 * w2;
  return h + bs + NOISE_C * z + ALPHA_C * (rec - h);
}

// One workgroup per batch row; 256 threads x 8 h-values each; state in registers.
__global__ __launch_bounds__(256) void rnn_scan(
    const float* __restrict__ input, const float* __restrict__ noise,
    const float* __restrict__ ws, float* __restrict__ traj)
{
  __shared__ float xs[T_DIM * I_DIM];
  __shared__ float part[2][16];   // [t parity][wave*2 + rank]
  const int b = blockIdx.x;
  const int tid = threadIdx.x;
  for (int i = tid; i < T_DIM * I_DIM; i += 256)
    xs[i] = input[(size_t)b * (T_DIM * I_DIM) + i];

  const int hA = 4 * tid;           // h in [0,1024)
  const int hB = 1024 + 4 * tid;    // h in [1024,2048)
  v4f wa0 = ld4(ws + 0 * H_DIM + hA), wb0 = ld4(ws + 0 * H_DIM + hB);
  v4f wa1 = ld4(ws + 1 * H_DIM + hA), wb1 = ld4(ws + 1 * H_DIM + hB);
  v4f wa2 = ld4(ws + 2 * H_DIM + hA), wb2 = ld4(ws + 2 * H_DIM + hB);
  v4f n0a = ld4(ws + 3 * H_DIM + hA), n0b = ld4(ws + 3 * H_DIM + hB);
  v4f n1a = ld4(ws + 4 * H_DIM + hA), n1b = ld4(ws + 4 * H_DIM + hB);
  v4f m0a = ld4(ws + 5 * H_DIM + hA), m0b = ld4(ws + 5 * H_DIM + hB);
  v4f m1a = ld4(ws + 6 * H_DIM + hA), m1b = ld4(ws + 6 * H_DIM + hB);
  v4f bia = ld4(ws + 9 * H_DIM + hA), bib = ld4(ws + 9 * H_DIM + hB);
  v4f ha  = ld4(ws + 10 * H_DIM + hA), hb = ld4(ws + 10 * H_DIM + hB);
  __syncthreads();

  const int lane = tid & 31, wave = tid >> 5;
  for (int t = 0; t < T_DIM; ++t) {
    const size_t base = ((size_t)b * T_DIM + t) * H_DIM;
    v4f za = ld4nt(noise + base + hA);   // issued early, consumed after reduce
    v4f zb = ld4nt(noise + base + hB);

    v4f ra = tanh4(ha), rb = tanh4(hb);
    float p0 = dot4(ra, n0a) + dot4(rb, n0b);
    float p1 = dot4(ra, n1a) + dot4(rb, n1b);
#pragma unroll
    for (int off = 16; off > 0; off >>= 1) {
      p0 += __shfl_xor(p0, off, 32);
      p1 += __shfl_xor(p1, off, 32);
    }
    const int par = t & 1;
    if (lane == 0) { part[par][2 * wave] = p0; part[par][2 * wave + 1] = p1; }
    __syncthreads();   // single barrier per step (parity double-buffer)
    float s0 = 0.f, s1 = 0.f;
#pragma unroll
    for (int w = 0; w < 8; ++w) { s0 += part[par][2 * w]; s1 += part[par][2 * w + 1]; }

    const float x0 = xs[t * 3 + 0], x1 = xs[t * 3 + 1], x2 = xs[t * 3 + 2];
    ha = upd(ha, za, bia, m0a, m1a, wa0, wa1, wa2, s0, s1, x0, x1, x2);
    hb = upd(hb, zb, bib, m0b, m1b, wb0, wb1, wb2, s0, s1, x0, x1, x2);
    *(v4f*)(traj + base + hA) = ha;
    *(v4f*)(traj + base + hB) = hb;
  }
}

// out[b,t,:] = tanh(traj[b,t,:]) @ wo_full : (16384 x 2048) x (2048 x 2) GEMM.
// One 16-row M-tile per wave via V_WMMA_F32_16X16X4_F32 (f32 end-to-end).
__global__ __launch_bounds__(256) void out_proj(
    const float* __restrict__ traj, const float* __restrict__ ws,
    float* __restrict__ out)
{
#if defined(__has_builtin) && __has_builtin(__builtin_amdgcn_wmma_f32_16x16x4_f32)
  const int lane = threadIdx.x & 31;
  const int wave = threadIdx.x >> 5;
  const int tile = blockIdx.x * 8 + wave;       // 1024 tiles of 16 rows
  const int mrow = lane & 15;                   // A row / B column index
  const int half = lane >> 4;                   // K-pair selector
  // Per-lane loop-invariant B column pointer: n=0 -> wo0, n=1 -> wo1,
  // n>=2 -> zero filler column. Removes all per-iteration selects.
  const float* __restrict__ bcol =
      ws + (size_t)((mrow == 0) ? 7 : (mrow == 1) ? 8 : 11) * H_DIM;
  const float* __restrict__ trow = traj + ((size_t)tile * 16 + mrow) * H_DIM;

  v8f c = {0.f, 0.f, 0.f, 0.f, 0.f, 0.f, 0.f, 0.f};
#pragma unroll 4
  for (int k0 = 0; k0 < H_DIM; k0 += 4) {
    const int k = k0 + half * 2;
    const v2f tp = *(const v2f*)(trow + k);     // A pair: one b64 load per lane
    const v2f bf = *(const v2f*)(bcol + k);     // B pair: one b64 load per lane
    v2f a;                                      // A[m][k], A[m][k+1]
    a.x = tanh_fast(tp.x);
    a.y = tanh_fast(tp.y);
    c = __builtin_amdgcn_wmma_f32_16x16x4_f32(false, a, false, bf,
                                              (short)0, c, false, false);
  }
  if (mrow < O_DIM) {
#pragma unroll
    for (int j = 0; j < 8; ++j) {               // D: VGPR j -> row j + 8*half
      const int row = tile * 16 + j + half * 8;
      out[(size_t)row * O_DIM + mrow] = c[j];
    }
  }
#else
  // Fallback (also parsed on host pass): one row per thread.
  const int g = blockIdx.x * 256 + threadIdx.x;
  if (g < B_DIM * T_DIM) {
    const float* __restrict__ trow = traj + (size_t)g * H_DIM;
    const float* __restrict__ wo0 = ws + 7 * H_DIM;
    const float* __restrict__ wo1 = ws + 8 * H_DIM;
    float a0 = 0.f, a1 = 0.f;
    for (int k = 0; k < H_DIM; ++k) {
      const float r = tanh_fast(trow[k]);
      a0 += r * wo0[k];
      a1 += r * wo1[k];
    }
    out[(size_t)g * O_DIM + 0] = a0;
    out[(size_t)g * O_DIM + 1] = a1;
  }
#endif
}

extern "C" void kernel_launch(void* const* d_in, const int* in_sizes, int n_in,
                              void* d_out, int out_size, void* d_ws, size_t ws_size,
                              hipStream_t stream) {
  (void)in_sizes; (void)n_in; (void)out_size; (void)ws_size;
  const float* input  = (const float*)d_in[0];
  const float* noise  = (const float*)d_in[1];
  const float* wi     = (const float*)d_in[2];
  const float* unitwi = (const float*)d_in[3];
  const float* m      = (const float*)d_in[4];
  const float* n      = (const float*)d_in[5];
  const float* unitm  = (const float*)d_in[6];
  const float* unitn  = (const float*)d_in[7];
  const float* wo     = (const float*)d_in[8];
  const float* h0     = (const float*)d_in[9];
  const float* unith0 = (const float*)d_in[10];
  const float* bias   = (const float*)d_in[11];
  const float* gb     = (const float*)d_in[12];
  const float* uv     = (const float*)d_in[13];
  const float* sup    = (const float*)d_in[14];

  float* out  = (float*)d_out;                       // (B,T,O) first
  float* traj = out + (size_t)B_DIM * T_DIM * O_DIM; // then (B,T,H)
  float* ws   = (float*)d_ws;

  build_proxies<<<(H_DIM + 255) / 256, 256, 0, stream>>>(
      wi, unitwi, m, n, unitm, unitn, wo, h0, unith0, bias, gb, uv, sup, ws);
  rnn_scan<<<B_DIM, 256, 0, stream>>>(input, noise, ws, traj);
  out_proj<<<(B_DIM * T_DIM) / 16 / 8, 256, 0, stream>>>(traj, ws, out);
}